// AttentionHyperNet_44092134261294
// MI455X (gfx1250) — compile-verified
//
#include <hip/hip_runtime.h>

typedef __attribute__((ext_vector_type(2))) float v2f;
typedef __attribute__((ext_vector_type(4))) float v4f;
typedef __attribute__((ext_vector_type(8))) float v8f;
typedef __attribute__((ext_vector_type(4))) unsigned int u32x4;
typedef __attribute__((ext_vector_type(8))) int i32x8;
typedef __attribute__((ext_vector_type(4))) int i32x4;

constexpr int kBS = 2048;
constexpr int kNE = 64;
constexpr int kNA = 16;
constexpr int kED = 128;
constexpr int kH  = 256;
constexpr int kNH = 8;
constexpr int kHD = 32;
constexpr int kM  = 32;
constexpr float kNEG = -1e30f;

// LDS layout (float offsets). Regions reused across phases:
//   [oE]  entities 64x128 (phase1)  ->  [oQ] 16x256 + [oA] 16x256 afterwards
constexpr int oX1 = 0;                    // 64x256 x1
constexpr int oK  = oX1 + 64 * 256;       // 64x256 k
constexpr int oV  = oK  + 64 * 256;       // 64x256 v
constexpr int oE  = oV  + 64 * 256;       // 64x128 entities (phase 1 only)
constexpr int oQ  = oE;                   // 16x256 q  (aliases dead entities)
constexpr int oA  = oQ  + 16 * 256;       // 16x256 attn
constexpr int oAO = oA  + 16 * 256;       // 16x256 attn_out
constexpr int oL  = oAO + 16 * 256;       // 8 x 16 x 64 logits / weights
constexpr int kLdsFloats = oL + kNH * kNA * kNE;   // 69632 floats = 272 KB

__device__ __forceinline__ v8f wmma4(v2f a, v2f b, v8f c) {
  // D = A(16x4 f32) * B(4x16 f32) + C(16x16 f32)
  return __builtin_amdgcn_wmma_f32_16x16x4_f32(
      /*neg_a=*/false, a, /*neg_b=*/false, b,
      /*c_mod=*/(short)0, c, /*reuse_a=*/false, /*reuse_b=*/false);
}

__global__ void __launch_bounds__(256, 1)
attn_hyper_kernel(const float* __restrict__ ent, const int* __restrict__ mask,
                  const float* __restrict__ W1,  const float* __restrict__ b1,
                  const float* __restrict__ Win, const float* __restrict__ Wout,
                  const float* __restrict__ bout, const float* __restrict__ W2,
                  const float* __restrict__ b2,  float* __restrict__ out) {
  extern __shared__ float sm[];
  const int b    = blockIdx.x;
  const int tid  = threadIdx.x;
  const int lane = tid & 31;
  const int wave = tid >> 5;
  const int half = lane >> 4;   // which 16-lane half of the wave
  const int l16  = lane & 15;
  const int* emrow = mask + (size_t)b * kNE;

  // ---- Stage entities[b] (64x128 f32 = 32KB) into LDS via the Tensor Data
  // Mover: one DMA descriptor, issued by wave 0 only (TDM ignores EXEC; one
  // issue = one full tile copy). Tracked by TENSORcnt.
  if (wave == 0) {
    const unsigned long long ga =
        (unsigned long long)(const void*)(ent + (size_t)b * kNE * kED);
    // D# group 0: count=1 (valid), lds_addr, global_addr[56:0], type=2
    u32x4 g0;
    g0[0] = 1u;
    g0[1] = (unsigned)(oE * sizeof(float));
    g0[2] = (unsigned)(ga & 0xFFFFFFFFull);
    g0[3] = (unsigned)((ga >> 32) & 0x01FFFFFFull) | 0x80000000u;
    // D# group 1: data_size=4B; tensor_dim0=8192, tensor_dim1=1;
    // tile_dim0=8192, tile_dim1=1; tensor_dim0_stride=8192 (elements)
    i32x8 g1;
    g1[0] = 0x00020000;            // [17:16] data_size = 2 (4 bytes)
    g1[1] = (int)(8192u << 16);    // [63:48]  tensor_dim0[15:0]
    g1[2] = (int)(1u << 16);       // [79:64]=dim0 hi=0, [95:80] tensor_dim1=1
    g1[3] = (int)(8192u << 16);    // [111:96]=dim1 hi=0, [127:112] tile_dim0
    g1[4] = 1;                     // [143:128] tile_dim1=1, [159:144] tile_dim2=0
    g1[5] = 8192;                  // [191:160] tensor_dim0_stride lo
    g1[6] = (int)(8192u << 16);    // [207:192]=stride hi=0, [223:208] dim1_stride lo
    g1[7] = 0;                     // [255:224] dim1_stride hi
    i32x4 z4 = {};
    i32x8 z8 = {};
    __builtin_amdgcn_tensor_load_to_lds(g0, g1, z4, z4, z8, 0);
    __builtin_amdgcn_s_wait_tensorcnt(0);
  }
  __syncthreads();

  // ---- GEMM1: x1(64x256) = relu(E(64x128) * W1^T(128x256) + b1) ----
  for (int t = wave; t < 64; t += 8) {
    const int mb = (t >> 4) * 16;
    const int nb = (t & 15) * 16;
    v8f c = {};
    const float* arow = sm + oE + (mb + l16) * kED + half * 2;
    const float* brow = W1 + (size_t)(nb + l16) * kED + half * 2;
#pragma unroll 8
    for (int k = 0; k < kED; k += 4)
      c = wmma4(*(const v2f*)(arow + k), *(const v2f*)(brow + k), c);
#pragma unroll
    for (int r = 0; r < 8; ++r) {
      const int row = mb + r + half * 8;
      const int col = nb + l16;
      const float v = c[r] + b1[col];
      sm[oX1 + row * kH + col] = v > 0.f ? v : 0.f;
    }
  }
  __syncthreads();

  // ---- GEMM2: qkv = x1(64x256) * Win^T(256x768); Q only for 16 agent rows --
  for (int t = wave; t < 144; t += 8) {
    int mb, fb, col0, dsto;
    if (t < 16)      { mb = 0;                 fb = t * 16;              col0 = t * 16;        dsto = oQ; }
    else if (t < 80) { int u = t - 16; mb = (u >> 4) * 16; fb = 256 + (u & 15) * 16; col0 = (u & 15) * 16; dsto = oK; }
    else             { int u = t - 80; mb = (u >> 4) * 16; fb = 512 + (u & 15) * 16; col0 = (u & 15) * 16; dsto = oV; }
    v8f c = {};
    const float* arow = sm + oX1 + (mb + l16) * kH + half * 2;
    const float* brow = Win + (size_t)(fb + l16) * kH + half * 2;
#pragma unroll 8
    for (int k = 0; k < kH; k += 4)
      c = wmma4(*(const v2f*)(arow + k), *(const v2f*)(brow + k), c);
#pragma unroll
    for (int r = 0; r < 8; ++r) {
      const int row = mb + r + half * 8;
      sm[dsto + row * kH + col0 + l16] = c[r];
    }
  }
  __syncthreads();

  // ---- GEMM3: logits[h] = Q_h(16x32) * K_h^T(32x64) / sqrt(HD); head=wave --
  {
    const int h = wave;
    const float* arow = sm + oQ + l16 * kH + h * kHD + half * 2;
    const float sc = 0.17677669529663687f;  // 1/sqrt(32)
    for (int nt = 0; nt < 4; ++nt) {
      v8f c = {};
      const float* brow = sm + oK + (nt * 16 + l16) * kH + h * kHD + half * 2;
#pragma unroll
      for (int k = 0; k < kHD; k += 4)
        c = wmma4(*(const v2f*)(arow + k), *(const v2f*)(brow + k), c);
#pragma unroll
      for (int r = 0; r < 8; ++r) {
        const int row = r + half * 8;
        sm[oL + h * (kNA * kNE) + row * kNE + nt * 16 + l16] = c[r] * sc;
      }
    }
  }
  __syncthreads();

  // ---- Masked softmax over k (reference polarity: mask where am|em set) ----
  if (tid < kNH * kNA) {
    const int h = tid >> 4, q = tid & 15;
    float* row = sm + oL + h * (kNA * kNE) + q * kNE;
    const bool amq = emrow[q] != 0;
    float mx = kNEG;
    bool allm = true;
    for (int k = 0; k < kNE; ++k) {
      const bool mk = amq || (emrow[k] != 0);
      const float v = mk ? kNEG : row[k];
      row[k] = v;
      if (!mk) allm = false;
      mx = fmaxf(mx, v);
    }
    float s = 0.f;
    for (int k = 0; k < kNE; ++k) {
      const float e = __expf(row[k] - mx);
      row[k] = e;
      s += e;
    }
    const float inv = allm ? 0.f : (1.f / s);
    for (int k = 0; k < kNE; ++k) row[k] *= inv;
  }
  __syncthreads();

  // ---- GEMM4: attn_h(16x32) = w_h(16x64) * V_h(64x32) ----
  for (int t = wave; t < 16; t += 8) {
    const int h = t >> 1;
    const int n0 = (t & 1) * 16;
    const float* arow = sm + oL + h * (kNA * kNE) + l16 * kNE + half * 2;
    v8f c = {};
    for (int k = 0; k < kNE; k += 4) {
      const int kk = k + half * 2;
      v2f bb;
      bb.x = sm[oV + (kk)     * kH + h * kHD + n0 + l16];
      bb.y = sm[oV + (kk + 1) * kH + h * kHD + n0 + l16];
      c = wmma4(*(const v2f*)(arow + k), bb, c);
    }
#pragma unroll
    for (int r = 0; r < 8; ++r) {
      const int row = r + half * 8;
      sm[oA + row * kH + h * kHD + n0 + l16] = c[r];
    }
  }
  __syncthreads();

  // ---- GEMM5: attn_out(16x256) = attn * Wout^T + bout, zero where agent ----
  for (int t = wave; t < 16; t += 8) {
    const int n0 = t * 16;
    const float* arow = sm + oA + l16 * kH + half * 2;
    const float* brow = Wout + (size_t)(n0 + l16) * kH + half * 2;
    v8f c = {};
#pragma unroll 8
    for (int k = 0; k < kH; k += 4)
      c = wmma4(*(const v2f*)(arow + k), *(const v2f*)(brow + k), c);
#pragma unroll
    for (int r = 0; r < 8; ++r) {
      const int row = r + half * 8;
      const int col = n0 + l16;
      const bool ag = emrow[row] != 0;
      sm[oAO + row * kH + col] = ag ? 0.f : (c[r] + bout[col]);
    }
  }
  __syncthreads();

  // ---- GEMM6: x3(16x32) = attn_out * W2^T + b2, zero where agent -> global -
  if (wave < 2) {
    const int n0 = wave * 16;
    const float* arow = sm + oAO + l16 * kH + half * 2;
    const float* brow = W2 + (size_t)(n0 + l16) * kH + half * 2;
    v8f c = {};
#pragma unroll 8
    for (int k = 0; k < kH; k += 4)
      c = wmma4(*(const v2f*)(arow + k), *(const v2f*)(brow + k), c);
#pragma unroll
    for (int r = 0; r < 8; ++r) {
      const int row = r + half * 8;
      const int col = n0 + l16;
      const bool ag = emrow[row] != 0;
      out[((size_t)b * kNA + row) * kM + col] = ag ? 0.f : (c[r] + b2[col]);
    }
  }
}

extern "C" void kernel_launch(void* const* d_in, const int* in_sizes, int n_in,
                              void* d_out, int out_size, void* d_ws, size_t ws_size,
                              hipStream_t stream) {
  const float* ent  = (const float*)d_in[0];
  const int*   msk  = (const int*)d_in[1];
  const float* W1   = (const float*)d_in[2];
  const float* b1   = (const float*)d_in[3];
  const float* Win  = (const float*)d_in[4];
  const float* Wout = (const float*)d_in[5];
  const float* bout = (const float*)d_in[6];
  const float* W2   = (const float*)d_in[7];
  const float* b2   = (const float*)d_in[8];
  float* out = (float*)d_out;

  const size_t shmem = (size_t)kLdsFloats * sizeof(float);  // 272 KB (<320 KB/WGP)
  hipLaunchKernelGGL(attn_hyper_kernel, dim3(kBS), dim3(256), shmem, stream,
                     ent, msk, W1, b1, Win, Wout, bout, W2, b2, out);
}